// BernNetHead_75780402970864
// MI455X (gfx1250) — compile-verified
//
#include <hip/hip_runtime.h>

typedef float v2f __attribute__((ext_vector_type(2)));
typedef float v8f __attribute__((ext_vector_type(8)));
typedef int   v4i __attribute__((ext_vector_type(4)));

#define N_NODES 1024
#define F0_DIM  256
#define HID     128
#define NB      4
#define KORD    8
#define ROWS    (NB * N_NODES)   // 4096

// B-tile pair-interleaved layout: BtP[kpair][c][2], kpair stride padded so
// (stride % 64 dwords) == 32 -> lanes 16..31 hit the opposite LDS bank half.
#define KP_STRIDE 288            // floats per kpair row (128*2 + 32 pad)

// ---------------------------------------------------------------------------
// Async global->LDS staging (gfx1250 GLOBAL_LOAD_ASYNC_TO_LDS_B128, ASYNCcnt)
// ---------------------------------------------------------------------------
#if defined(__has_builtin)
#if __has_builtin(__builtin_amdgcn_global_load_async_to_lds_b128) && \
    __has_builtin(__builtin_amdgcn_s_wait_asynccnt)
#define HAVE_ASYNC_LDS 1
#endif
#endif

#define AS3 __attribute__((address_space(3)))

#ifdef HAVE_ASYNC_LDS
__device__ __forceinline__ void stage16(const float* g, float* l) {
  __builtin_amdgcn_global_load_async_to_lds_b128((v4i*)g, (AS3 v4i*)l, 0, 0);
}
__device__ __forceinline__ void stage_wait() {
  __builtin_amdgcn_s_wait_asynccnt(0);
}
#else
__device__ __forceinline__ void stage16(const float* g, float* l) {
  *reinterpret_cast<float4*>(l) = *reinterpret_cast<const float4*>(g);
}
__device__ __forceinline__ void stage_wait() {}
#endif

// ---------------------------------------------------------------------------
// Shared helpers
// ---------------------------------------------------------------------------
__device__ inline float blockReduceSum(float v, float* sm) {
  int t = threadIdx.x;
  sm[t] = v;
  __syncthreads();
  for (int s = 128; s > 0; s >>= 1) {
    if (t < s) sm[t] += sm[t + s];
    __syncthreads();
  }
  float r = sm[0];
  __syncthreads();
  return r;
}

// Inner WMMA micro-kernel for one 16-wide K chunk:
// preload 4 A-frags + 8 B-frags per k-step, then issue 8 WMMAs back-to-back.
__device__ __forceinline__ void wmma_chunk(const float* At, const float* BtP,
                                           int mr, int lh, int l15, v8f acc[8]) {
  v2f afr[4];
#pragma unroll
  for (int kk = 0; kk < 4; ++kk)
    afr[kk] = *reinterpret_cast<const v2f*>(&At[mr * 20 + kk * 4 + lh * 2]);
#pragma unroll
  for (int kk = 0; kk < 4; ++kk) {
    const int kp = (kk * 4 + lh * 2) >> 1;
    v2f bf[8];
#pragma unroll
    for (int j = 0; j < 8; ++j)
      bf[j] = *reinterpret_cast<const v2f*>(&BtP[kp * KP_STRIDE + (j * 16 + l15) * 2]);
#pragma unroll
    for (int j = 0; j < 8; ++j)
      acc[j] = __builtin_amdgcn_wmma_f32_16x16x4_f32(
          false, afr[kk], false, bf[j], (short)0, acc[j], false, false);
  }
}

// ---------------------------------------------------------------------------
// WMMA GEMM #1: Out[r,h] = sum_f X[r,f] * W[h,f] + bias[h]
//   X: [4096, 256] row-major, W: [128, 256] row-major, Out: [4096, 128]
//   grid.x = 32 (row tiles of 128), block = 256 threads (8 wave32)
// ---------------------------------------------------------------------------
__global__ __launch_bounds__(256) void gemm_xw_kernel(
    const float* __restrict__ X, const float* __restrict__ W,
    const float* __restrict__ bias, float* __restrict__ Out) {
  __shared__ float At[128 * 20];          // A tile [row][k], stride 20
  __shared__ float BtP[8 * KP_STRIDE];    // B tile pair-interleaved [kpair][h][2]
  const int t = threadIdx.x;
  const int wave = t >> 5, lane = t & 31;
  const int lh = lane >> 4, l15 = lane & 15;
  const int bRow = blockIdx.x * 128;
  v8f zero = {};
  v8f acc[8];
#pragma unroll
  for (int j = 0; j < 8; ++j) acc[j] = zero;

  for (int kb = 0; kb < F0_DIM; kb += 16) {
#pragma unroll
    for (int i = 0; i < 2; ++i) {                 // A tile: 128x16, async b128
      int idx = t + i * 256;
      int r = idx >> 2, kq = (idx & 3) << 2;
      stage16(&X[(size_t)(bRow + r) * F0_DIM + kb + kq], &At[r * 20 + kq]);
    }
#pragma unroll
    for (int i = 0; i < 2; ++i) {                 // W tile -> pair-interleaved
      int idx = t + i * 256;
      int h = idx >> 2, kq = (idx & 3) << 2;
      float4 v = *reinterpret_cast<const float4*>(&W[(size_t)h * F0_DIM + kb + kq]);
      BtP[((kq + 0) >> 1) * KP_STRIDE + h * 2 + ((kq + 0) & 1)] = v.x;
      BtP[((kq + 1) >> 1) * KP_STRIDE + h * 2 + ((kq + 1) & 1)] = v.y;
      BtP[((kq + 2) >> 1) * KP_STRIDE + h * 2 + ((kq + 2) & 1)] = v.z;
      BtP[((kq + 3) >> 1) * KP_STRIDE + h * 2 + ((kq + 3) & 1)] = v.w;
    }
    stage_wait();
    __syncthreads();
    wmma_chunk(At, BtP, wave * 16 + l15, lh, l15, acc);
    __syncthreads();
  }
#pragma unroll
  for (int j = 0; j < 8; ++j) {
    int col = j * 16 + l15;
    float bv = bias[col];
#pragma unroll
    for (int v = 0; v < 8; ++v) {                 // C frag: M=v+lh*8, N=l15
      int row = bRow + wave * 16 + lh * 8 + v;
      Out[(size_t)row * HID + col] = acc[j][v] + bv;
    }
  }
}

// ---------------------------------------------------------------------------
// WMMA GEMM #2: Out[b] = L @ In[b]  (per-batch node-space operator)
//   L: [1024, 1024], In/Out: [B, 1024, 128]
//   grid = (8, 1, 4), block = 256 threads
// ---------------------------------------------------------------------------
__global__ __launch_bounds__(256) void gemm_L_kernel(
    const float* __restrict__ L, const float* __restrict__ In,
    float* __restrict__ Out) {
  __shared__ float At[128 * 20];          // L tile [row][k]
  __shared__ float BtP[8 * KP_STRIDE];    // X tile pair-interleaved [kpair][c][2]
  const int t = threadIdx.x;
  const int wave = t >> 5, lane = t & 31;
  const int lh = lane >> 4, l15 = lane & 15;
  const int bRow = blockIdx.x * 128;
  const float* X = In + (size_t)blockIdx.z * N_NODES * HID;
  float* O = Out + (size_t)blockIdx.z * N_NODES * HID;
  v8f zero = {};
  v8f acc[8];
#pragma unroll
  for (int j = 0; j < 8; ++j) acc[j] = zero;

  for (int kb = 0; kb < N_NODES; kb += 16) {
    if (kb + 16 < N_NODES)   // hint next L chunk (global_prefetch_b8)
      __builtin_prefetch(&L[(size_t)bRow * N_NODES + kb + 16], 0, 1);
#pragma unroll
    for (int i = 0; i < 2; ++i) {                 // L tile: 128x16, async b128
      int idx = t + i * 256;
      int r = idx >> 2, kq = (idx & 3) << 2;
      stage16(&L[(size_t)(bRow + r) * N_NODES + kb + kq], &At[r * 20 + kq]);
    }
#pragma unroll
    for (int i = 0; i < 2; ++i) {                 // X tile -> pair-interleaved
      int idx = t + i * 256;
      int r = idx >> 5, cq = (idx & 31) << 2;     // r: k row 0..15, cq: col quad
      float4 v = *reinterpret_cast<const float4*>(&X[(size_t)(kb + r) * HID + cq]);
      float* base = &BtP[(r >> 1) * KP_STRIDE + (r & 1)];
      base[(cq + 0) * 2] = v.x;
      base[(cq + 1) * 2] = v.y;
      base[(cq + 2) * 2] = v.z;
      base[(cq + 3) * 2] = v.w;
    }
    stage_wait();
    __syncthreads();
    wmma_chunk(At, BtP, wave * 16 + l15, lh, l15, acc);
    __syncthreads();
  }
#pragma unroll
  for (int j = 0; j < 8; ++j) {
    int col = j * 16 + l15;
#pragma unroll
    for (int v = 0; v < 8; ++v) {
      int row = bRow + wave * 16 + lh * 8 + v;
      O[(size_t)row * HID + col] = acc[j][v];
    }
  }
}

// ---------------------------------------------------------------------------
// BatchNorm statistics (biased variance, torch training semantics)
// ---------------------------------------------------------------------------
__global__ __launch_bounds__(256) void bn_stats_kernel(
    const float* __restrict__ x, float* __restrict__ mu, float* __restrict__ rstd) {
  __shared__ float sm[256];
  int c = blockIdx.x;
  float s = 0.f, ss = 0.f;
  for (int r = threadIdx.x; r < ROWS; r += 256) {
    float v = x[(size_t)r * HID + c];
    s += v; ss += v * v;
  }
  s = blockReduceSum(s, sm);
  ss = blockReduceSum(ss, sm);
  if (threadIdx.x == 0) {
    float m = s * (1.f / ROWS);
    float var = ss * (1.f / ROWS) - m * m;
    mu[c] = m;
    rstd[c] = rsqrtf(var + 1e-5f);
  }
}

// y = (x - mu) * rstd * g + b  [+ relu]; grid = 2048, block = 256
__global__ __launch_bounds__(256) void bn_apply_kernel(
    float* __restrict__ y, const float* __restrict__ x,
    const float* __restrict__ mu, const float* __restrict__ rstd,
    const float* __restrict__ g, const float* __restrict__ b, int do_relu) {
  size_t i = (size_t)blockIdx.x * 256 + threadIdx.x;
  int h = (int)(i & (HID - 1));
  float v = (x[i] - mu[h]) * rstd[h] * g[h] + b[h];
  if (do_relu) v = fmaxf(v, 0.f);
  y[i] = v;
}

// T_j = T_{j-1} - (L @ T_{j-1})
__global__ __launch_bounds__(256) void sub_kernel(
    float* __restrict__ Tj, const float* __restrict__ Tjm1,
    const float* __restrict__ LT) {
  size_t i = (size_t)blockIdx.x * 256 + threadIdx.x;
  Tj[i] = Tjm1[i] - LT[i];
}

// A = binom * theta_row[h] * T0
__global__ __launch_bounds__(256) void scale_kernel(
    float* __restrict__ A, const float* __restrict__ T0,
    const float* __restrict__ th, float binom) {
  size_t i = (size_t)blockIdx.x * 256 + threadIdx.x;
  int h = (int)(i & (HID - 1));
  A[i] = binom * th[h] * T0[i];
}

// A = At + binom * theta_row[h] * Tj   (Horner step on node operator)
__global__ __launch_bounds__(256) void horner_kernel(
    float* __restrict__ A, const float* __restrict__ At,
    const float* __restrict__ Tj, const float* __restrict__ th, float binom) {
  size_t i = (size_t)blockIdx.x * 256 + threadIdx.x;
  int h = (int)(i & (HID - 1));
  A[i] = At[i] + binom * th[h] * Tj[i];
}

// out[b*HID+h] = mean_n of (x2 ? relu(x1+x2) : x1); grid = 512, block = 256
__global__ __launch_bounds__(256) void mean_nodes_kernel(
    const float* __restrict__ x1, const float* __restrict__ x2,
    float* __restrict__ out) {
  __shared__ float sm[256];
  int bh = blockIdx.x;
  int b = bh >> 7, h = bh & (HID - 1);
  size_t base = (size_t)b * N_NODES * HID + h;
  float s = 0.f;
  for (int n = threadIdx.x; n < N_NODES; n += 256) {
    float v = x1[base + (size_t)n * HID];
    if (x2) { v += x2[base + (size_t)n * HID]; v = fmaxf(v, 0.f); }
    s += v;
  }
  s = blockReduceSum(s, sm);
  if (threadIdx.x == 0) out[bh] = s * (1.f / N_NODES);
}

// Final head: H_hidden = 0.5*(P+Zp) [4x128]; logits [4x10]
__global__ __launch_bounds__(576) void final_head_kernel(
    const float* __restrict__ P, const float* __restrict__ Zp,
    const float* __restrict__ fc2w, const float* __restrict__ fc2b,
    const float* __restrict__ m2w, const float* __restrict__ m2b,
    float* __restrict__ out) {
  int i = threadIdx.x;
  if (i < NB * HID) {
    out[i] = 0.5f * (P[i] + Zp[i]);
  } else if (i < NB * HID + NB * 10) {
    int j = i - NB * HID;
    int b = j / 10, o = j % 10;
    float s = fc2b[o] + m2b[o];
    for (int h = 0; h < HID; ++h)
      s += P[b * HID + h] * fc2w[o * HID + h] + Zp[b * HID + h] * m2w[o * HID + h];
    out[NB * HID + j] = s;
  }
}

// ---------------------------------------------------------------------------
// Launch
// ---------------------------------------------------------------------------
extern "C" void kernel_launch(void* const* d_in, const int* in_sizes, int n_in,
                              void* d_out, int out_size, void* d_ws, size_t ws_size,
                              hipStream_t stream) {
  const float* x     = (const float*)d_in[0];
  const float* L     = (const float*)d_in[1];
  const float* fc1w  = (const float*)d_in[2];
  const float* fc1b  = (const float*)d_in[3];
  const float* bn1g  = (const float*)d_in[4];
  const float* bn1b  = (const float*)d_in[5];
  const float* bn2g  = (const float*)d_in[6];
  const float* bn2b  = (const float*)d_in[7];
  const float* theta = (const float*)d_in[8];
  const float* fc2w  = (const float*)d_in[9];
  const float* fc2b  = (const float*)d_in[10];
  const float* m1w   = (const float*)d_in[11];
  const float* m1b   = (const float*)d_in[12];
  const float* mbng  = (const float*)d_in[13];
  const float* mbnb  = (const float*)d_in[14];
  const float* m2w   = (const float*)d_in[15];
  const float* m2b   = (const float*)d_in[16];

  const size_t SZ = (size_t)ROWS * HID;   // 524288 floats
  float* ws   = (float*)d_ws;
  float* H0   = ws + 0 * SZ;              // fc1 pre-BN
  float* Hn   = ws + 1 * SZ;              // post-bn1 H  (== T[0])
  float* T[KORD + 1];
  T[0] = Hn;
  for (int j = 1; j <= KORD; ++j) T[j] = ws + (1 + j) * SZ;   // offsets 2..9
  float* LT   = ws + 10 * SZ;
  float* A    = ws + 11 * SZ;
  float* At   = ws + 12 * SZ;
  float* Zm0  = ws + 13 * SZ;
  float* Zmn  = ws + 14 * SZ;
  float* Hfbn = ws + 15 * SZ;
  float* smallbuf = ws + 16 * SZ;
  float* mu1 = smallbuf + 0;
  float* rs1 = smallbuf + 128;
  float* mu2 = smallbuf + 256;
  float* rs2 = smallbuf + 384;
  float* mum = smallbuf + 512;
  float* rsm = smallbuf + 640;
  float* P   = smallbuf + 1024;
  float* Zp  = smallbuf + 1536;

  static const float BINOM[KORD + 1] = {1.f, 8.f, 28.f, 56.f, 70.f, 56.f, 28.f, 8.f, 1.f};

  const dim3 blk(256);
  const dim3 ew(2048);         // 2048*256 == ROWS*HID elementwise

  // fc1 and mlp1 streams (WMMA GEMMs)
  gemm_xw_kernel<<<dim3(32), blk, 0, stream>>>(x, fc1w, fc1b, H0);
  gemm_xw_kernel<<<dim3(32), blk, 0, stream>>>(x, m1w, m1b, Zm0);

  // bn1 on GCN stream
  bn_stats_kernel<<<dim3(128), blk, 0, stream>>>(H0, mu1, rs1);
  bn_apply_kernel<<<ew, blk, 0, stream>>>(Hn, H0, mu1, rs1, bn1g, bn1b, 0);

  // mlp bn + relu
  bn_stats_kernel<<<dim3(128), blk, 0, stream>>>(Zm0, mum, rsm);
  bn_apply_kernel<<<ew, blk, 0, stream>>>(Zmn, Zm0, mum, rsm, mbng, mbnb, 1);

  // T_j = (I-L)^j H via T_j = T_{j-1} - L*T_{j-1}
  for (int j = 1; j <= KORD; ++j) {
    gemm_L_kernel<<<dim3(8, 1, NB), blk, 0, stream>>>(L, T[j - 1], LT);
    sub_kernel<<<ew, blk, 0, stream>>>(T[j], T[j - 1], LT);
  }

  // Horner: A = sum_k binom[k]*theta[k] (.) L^k T_{K-k}
  scale_kernel<<<ew, blk, 0, stream>>>(A, T[0], theta + KORD * HID, BINOM[KORD]);
  for (int k = KORD - 1; k >= 0; --k) {
    gemm_L_kernel<<<dim3(8, 1, NB), blk, 0, stream>>>(L, A, At);
    horner_kernel<<<ew, blk, 0, stream>>>(A, At, T[KORD - k], theta + k * HID, BINOM[k]);
  }

  // bn2 on Hf
  bn_stats_kernel<<<dim3(128), blk, 0, stream>>>(A, mu2, rs2);
  bn_apply_kernel<<<ew, blk, 0, stream>>>(Hfbn, A, mu2, rs2, bn2g, bn2b, 0);

  // pooled features: P = mean_n relu(Hn + Hfbn), Zp = mean_n Zmn
  mean_nodes_kernel<<<dim3(512), blk, 0, stream>>>(Hn, Hfbn, P);
  mean_nodes_kernel<<<dim3(512), blk, 0, stream>>>(Zmn, nullptr, Zp);

  // final head -> d_out = [H_hidden (512) | logits (40)]
  final_head_kernel<<<dim3(1), dim3(576), 0, stream>>>(P, Zp, fc2w, fc2b, m2w, m2b,
                                                       (float*)d_out);
}